// MRConv_86517821214608
// MI455X (gfx1250) — compile-verified
//
#include <hip/hip_runtime.h>

typedef __attribute__((ext_vector_type(2))) float v2f;
typedef __attribute__((ext_vector_type(8))) float v8f;

#define C_IN  64
#define C2    128
#define C_OUT 64

// Order-preserving float <-> int bijection so we can use native
// global_atomic_max_i32 for the segment-max (deterministic: max is
// order-independent, no float rounding in the atomic).
__device__ __forceinline__ int f2ord(float f) {
    int i = __float_as_int(f);
    return (i >= 0) ? i : (i ^ 0x7fffffff);
}
__device__ __forceinline__ float ord2f(int i) {
    int b = (i >= 0) ? i : (i ^ 0x7fffffff);
    return __int_as_float(b);
}

// f2ord(-inf) == 0x807fffff : smallest encoding we ever produce.
__global__ void mrconv_init_kernel(int* __restrict__ ord, long long n) {
    long long i = (long long)blockIdx.x * blockDim.x + threadIdx.x;
    if (i < n) ord[i] = (int)0x807fffffu;
}

// One wave32 per edge; lane owns channels 2*lane, 2*lane+1 (float2 loads).
// x (25.6 MB) and the atomic region are L2-resident on the 192 MB L2;
// native int32 atomic max is the throughput-limiting op.
__global__ __launch_bounds__(256) void mrconv_scatter_max_kernel(
        const float* __restrict__ x,
        const int*   __restrict__ src,
        const int*   __restrict__ dst,
        int*         __restrict__ ord,
        long long E) {
    long long gid = (long long)blockIdx.x * blockDim.x + threadIdx.x;
    long long e = gid >> 5;
    int lane = (int)(gid & 31);
    if (e >= E) return;
    size_t s = (size_t)src[e];
    size_t d = (size_t)dst[e];
    float2 a = *(const float2*)(x + s * C_IN + 2 * lane);
    float2 c = *(const float2*)(x + d * C_IN + 2 * lane);
    int* o = ord + d * C_IN + 2 * lane;
    atomicMax(&o[0], f2ord(a.x - c.x));
    atomicMax(&o[1], f2ord(a.y - c.y));
}

// out[N,64] = relu([x | xj][N,128] @ W[128,64] + b)  via V_WMMA_F32_16X16X4_F32.
// One wave per 16-row tile; 4 accumulators cover all 64 output columns.
// W is repacked in LDS as Wp[kpair][col][2] so each lane's B fragment
// (W[k0][col], W[k0+1][col]) is a single aligned b64 LDS load, and the four
// n-tile fragments share one base with byte offsets 0/128/256/384
// (-> ds_load_2addr_b64, results land directly in even-aligned VGPR pairs).
__global__ __launch_bounds__(256) void mrconv_gemm_wmma_kernel(
        const float* __restrict__ x,
        const int*   __restrict__ ordxj,
        const float* __restrict__ W,
        const float* __restrict__ bias,
        float*       __restrict__ out,
        int N) {
    __shared__ float Wp[C2 * C_OUT];              // 32 KB, packed layout
    for (int i = threadIdx.x; i < C2 * C_OUT; i += 256) {
        int k   = i / C_OUT;
        int col = i - k * C_OUT;
        Wp[(k >> 1) * (2 * C_OUT) + col * 2 + (k & 1)] = W[i];
    }
    __syncthreads();

    int wave  = threadIdx.x >> 5;
    int lane  = threadIdx.x & 31;
    int m0    = (blockIdx.x * 8 + wave) * 16;     // 16-row tile base
    if (m0 >= N) return;
    int mrow  = lane & 15;                        // M (and N) position of lane
    int khalf = lane >> 4;                        // K half: lanes 16-31 hold K+2..K+3
    int row   = m0 + mrow;
    if (row >= N) row = N - 1;                    // safety; stores are guarded

    const float* xrow = x     + (size_t)row * C_IN;
    const int*   jrow = ordxj + (size_t)row * C_IN;
    const v2f*   wcol = (const v2f*)&Wp[mrow * 2];  // + kpair*64 v2f elements

    v8f acc0 = {}, acc1 = {}, acc2 = {}, acc3 = {};

    // K = 0..63 : A from x
    #pragma unroll
    for (int kb = 0; kb < C_IN; kb += 4) {
        int k0 = kb + khalf * 2;                  // this lane's K (and K+1)
        v2f a;  a.x = xrow[k0];  a.y = xrow[k0 + 1];
        const v2f* bb = wcol + (size_t)(k0 >> 1) * C_OUT;
        v2f b0 = bb[0];
        v2f b1 = bb[16];
        v2f b2 = bb[32];
        v2f b3 = bb[48];
        acc0 = __builtin_amdgcn_wmma_f32_16x16x4_f32(false, a, false, b0, (short)0, acc0, false, false);
        acc1 = __builtin_amdgcn_wmma_f32_16x16x4_f32(false, a, false, b1, (short)0, acc1, false, false);
        acc2 = __builtin_amdgcn_wmma_f32_16x16x4_f32(false, a, false, b2, (short)0, acc2, false, false);
        acc3 = __builtin_amdgcn_wmma_f32_16x16x4_f32(false, a, false, b3, (short)0, acc3, false, false);
    }
    // K = 64..127 : A from decoded segment-max (empty/underflow segments -> 0)
    #pragma unroll
    for (int kb = 0; kb < C_IN; kb += 4) {
        int k0 = kb + khalf * 2;
        float a0 = ord2f(jrow[k0]);     if (a0 < -10000.0f) a0 = 0.0f;
        float a1 = ord2f(jrow[k0 + 1]); if (a1 < -10000.0f) a1 = 0.0f;
        v2f a;  a.x = a0;  a.y = a1;
        const v2f* bb = wcol + (size_t)((C_IN + k0) >> 1) * C_OUT;
        v2f b0 = bb[0];
        v2f b1 = bb[16];
        v2f b2 = bb[32];
        v2f b3 = bb[48];
        acc0 = __builtin_amdgcn_wmma_f32_16x16x4_f32(false, a, false, b0, (short)0, acc0, false, false);
        acc1 = __builtin_amdgcn_wmma_f32_16x16x4_f32(false, a, false, b1, (short)0, acc1, false, false);
        acc2 = __builtin_amdgcn_wmma_f32_16x16x4_f32(false, a, false, b2, (short)0, acc2, false, false);
        acc3 = __builtin_amdgcn_wmma_f32_16x16x4_f32(false, a, false, b3, (short)0, acc3, false, false);
    }

    // C/D layout: VGPR r holds row M = khalf*8 + r, column N = mrow (per tile).
    #pragma unroll
    for (int nt = 0; nt < 4; ++nt) {
        v8f acc = (nt == 0) ? acc0 : (nt == 1) ? acc1 : (nt == 2) ? acc2 : acc3;
        int col = nt * 16 + mrow;
        float bb = bias[col];
        #pragma unroll
        for (int r = 0; r < 8; ++r) {
            int ro = m0 + khalf * 8 + r;
            if (ro < N) {
                float v = acc[r] + bb;
                out[(size_t)ro * C_OUT + col] = v > 0.0f ? v : 0.0f;
            }
        }
    }
}

extern "C" void kernel_launch(void* const* d_in, const int* in_sizes, int n_in,
                              void* d_out, int out_size, void* d_ws, size_t ws_size,
                              hipStream_t stream) {
    const float* x    = (const float*)d_in[0];
    const int*   ei   = (const int*)d_in[1];   // edge_index, [2, E]
    const float* W    = (const float*)d_in[2];
    const float* bias = (const float*)d_in[3];
    float*       out  = (float*)d_out;

    long long N = (long long)in_sizes[0] / C_IN;
    long long E = (long long)in_sizes[1] / 2;
    const int* src = ei;
    const int* dst = ei + E;
    int* ord = (int*)d_ws;                      // N*64 int32 = 25.6 MB scratch

    long long ntot = N * C_IN;
    mrconv_init_kernel<<<(unsigned)((ntot + 255) / 256), 256, 0, stream>>>(ord, ntot);

    long long sthreads = E * 32;                // one wave32 per edge
    mrconv_scatter_max_kernel<<<(unsigned)((sthreads + 255) / 256), 256, 0, stream>>>(
        x, src, dst, ord, E);

    int tiles  = (int)((N + 15) / 16);
    int blocks = (tiles + 7) / 8;               // 8 waves (tiles) per block
    mrconv_gemm_wmma_kernel<<<blocks, 256, 0, stream>>>(x, ord, W, bias, out, (int)N);
}